// HistogramLoss_80625126080915
// MI455X (gfx1250) — compile-verified
//
#include <hip/hip_runtime.h>
#include <stdint.h>

#define NUM_BINS 256
#define NUM_SLICES 48            // B*C = 16*3
#define SLICE_ELEMS (512 * 512)  // 262144 elements per (b,c) slice
#define BLOCKS_PER_SLICE 16
#define THREADS 256
#define WAVES (THREADS / 32)                        // 8 waves (wave32)
#define ELEMS_PER_BLOCK (SLICE_ELEMS / BLOCKS_PER_SLICE)  // 16384
#define VEC4_PER_BLOCK (ELEMS_PER_BLOCK / 4)              // 4096
#define VEC4_PER_THREAD (VEC4_PER_BLOCK / THREADS)        // 16

typedef float v2f __attribute__((ext_vector_type(2)));
typedef float v8f __attribute__((ext_vector_type(8)));

// ---------------------------------------------------------------------------
// Kernel 0: zero the per-slice histograms in workspace (graph replays must be
// deterministic; harness poisons ws with 0xAA).
// ---------------------------------------------------------------------------
__global__ void hist_zero_kernel(uint32_t* __restrict__ ws, int n) {
    int i = blockIdx.x * blockDim.x + threadIdx.x;
    if (i < n) ws[i] = 0u;
}

// ---------------------------------------------------------------------------
// Kernel 1: binning. grid = (48*16, 2); blockIdx.y selects pred/target.
// Per-wave private LDS histograms -> ds_add_u32 with minimal contention,
// then one global_atomic_add_u32 per (thread, bin) merge.
// ---------------------------------------------------------------------------
__global__ __launch_bounds__(THREADS)
void hist_accum_kernel(const float* __restrict__ pred,
                       const float* __restrict__ targ,
                       uint32_t* __restrict__ hist) {
    __shared__ uint32_t lh[WAVES][NUM_BINS];  // 8 KB

    const int slice = blockIdx.x / BLOCKS_PER_SLICE;
    const int chunk = blockIdx.x % BLOCKS_PER_SLICE;
    const float* __restrict__ src = (blockIdx.y == 0) ? pred : targ;
    uint32_t* __restrict__ hout =
        hist + ((size_t)blockIdx.y * NUM_SLICES + (size_t)slice) * NUM_BINS;

    const int t = threadIdx.x;
    const int wave = t >> 5;

    // Zero LDS histograms.
    uint32_t* lflat = &lh[0][0];
    for (int i = t; i < WAVES * NUM_BINS; i += THREADS) lflat[i] = 0u;
    __syncthreads();

    const float4* __restrict__ base =
        (const float4*)src + (size_t)slice * (SLICE_ELEMS / 4) +
        (size_t)chunk * VEC4_PER_BLOCK;

#pragma unroll
    for (int i = 0; i < VEC4_PER_THREAD; ++i) {
        if (i + 1 < VEC4_PER_THREAD)
            __builtin_prefetch(base + (i + 1) * THREADS + t, 0, 1);
        float4 v = base[i * THREADS + t];

        int b0 = (int)(v.x * 255.0f);
        int b1 = (int)(v.y * 255.0f);
        int b2 = (int)(v.z * 255.0f);
        int b3 = (int)(v.w * 255.0f);
        b0 = b0 < 0 ? 0 : (b0 > 255 ? 255 : b0);
        b1 = b1 < 0 ? 0 : (b1 > 255 ? 255 : b1);
        b2 = b2 < 0 ? 0 : (b2 > 255 ? 255 : b2);
        b3 = b3 < 0 ? 0 : (b3 > 255 ? 255 : b3);
        atomicAdd(&lh[wave][b0], 1u);
        atomicAdd(&lh[wave][b1], 1u);
        atomicAdd(&lh[wave][b2], 1u);
        atomicAdd(&lh[wave][b3], 1u);
    }
    __syncthreads();

    // Merge the 8 wave-private histograms; thread t owns bin t.
    uint32_t sum = 0;
#pragma unroll
    for (int w = 0; w < WAVES; ++w) sum += lh[w][t];
    atomicAdd(&hout[t], sum);
}

// ---------------------------------------------------------------------------
// Kernel 2: L1 reduction over 12288 bin pairs via V_WMMA_F32_16X16X4_F32.
// A(16x4) holds 64 |diff| values per step (2 per lane: lanes 0-15 -> K=0,1
// rows M=lane; lanes 16-31 -> K=2,3 rows M=lane-16 — any bijection works
// since B is all-ones). D[m][n] accumulates exact f32 row-sums.
// Single wave (32 threads) => EXEC all ones as WMMA requires.
// ---------------------------------------------------------------------------
__global__ __launch_bounds__(32)
void hist_l1_wmma_kernel(const uint32_t* __restrict__ hist,
                         float* __restrict__ out) {
    const int lane = threadIdx.x;
    const uint32_t* __restrict__ ph = hist;
    const uint32_t* __restrict__ th = hist + NUM_SLICES * NUM_BINS;

    const v2f ones = {1.0f, 1.0f};
    v8f acc0 = {}, acc1 = {}, acc2 = {}, acc3 = {};

    const int TOTAL = NUM_SLICES * NUM_BINS;  // 12288, divisible by 256
    for (int j = 0; j < TOTAL; j += 256) {
#pragma unroll
        for (int u = 0; u < 4; ++u) {
            const int i0 = j + u * 64 + 2 * lane;
            // counts <= 2^18 are exact in f32 -> exact integer |diff|
            float d0 = fabsf((float)(int)ph[i0] - (float)(int)th[i0]);
            float d1 = fabsf((float)(int)ph[i0 + 1] - (float)(int)th[i0 + 1]);
            v2f a = {d0, d1};
            if (u == 0)
                acc0 = __builtin_amdgcn_wmma_f32_16x16x4_f32(
                    false, a, false, ones, (short)0, acc0, false, false);
            else if (u == 1)
                acc1 = __builtin_amdgcn_wmma_f32_16x16x4_f32(
                    false, a, false, ones, (short)0, acc1, false, false);
            else if (u == 2)
                acc2 = __builtin_amdgcn_wmma_f32_16x16x4_f32(
                    false, a, false, ones, (short)0, acc2, false, false);
            else
                acc3 = __builtin_amdgcn_wmma_f32_16x16x4_f32(
                    false, a, false, ones, (short)0, acc3, false, false);
        }
    }

    v8f acc = acc0 + acc1 + acc2 + acc3;
    // Lane n holds C[0..7][n] (n<16) or C[8..15][n-16]; all columns identical.
    float s = acc[0] + acc[1] + acc[2] + acc[3] +
              acc[4] + acc[5] + acc[6] + acc[7];
    float total = __shfl(s, 0, 32) + __shfl(s, 16, 32);

    if (lane == 0)
        out[0] = total / ((float)SLICE_ELEMS * (float)TOTAL);
}

// ---------------------------------------------------------------------------
extern "C" void kernel_launch(void* const* d_in, const int* in_sizes, int n_in,
                              void* d_out, int out_size, void* d_ws,
                              size_t ws_size, hipStream_t stream) {
    const float* pred = (const float*)d_in[0];
    const float* targ = (const float*)d_in[1];
    uint32_t* hist = (uint32_t*)d_ws;  // 2 * 48 * 256 u32 = 96 KB
    float* out = (float*)d_out;

    const int nhist = 2 * NUM_SLICES * NUM_BINS;  // 24576
    hist_zero_kernel<<<(nhist + THREADS - 1) / THREADS, THREADS, 0, stream>>>(
        hist, nhist);

    dim3 grid(NUM_SLICES * BLOCKS_PER_SLICE, 2);
    hist_accum_kernel<<<grid, THREADS, 0, stream>>>(pred, targ, hist);

    hist_l1_wmma_kernel<<<1, 32, 0, stream>>>(hist, out);
}